// SingleNet_38963943310048
// MI455X (gfx1250) — compile-verified
//
#include <hip/hip_runtime.h>

// SingleNet meta-learning step for MI455X (gfx1250, wave32).
//
// D = 2048, MH = 32.
// Forward: 3x GEMV (+bias, ReLU) -- memory bound, b128 loads + wave32 butterfly.
// Meta:    per-weight MLP 3->32->1 over 12.6M entries -- compute bound.
//          K=4 contraction (3 inputs + bias) on V_WMMA_F32_16X16X4_F32,
//          transposed mapping: A = meta matrix (loop-invariant), B = entry
//          data. C holds H^T (meta rows across VGPRs/halves, entries across
//          lanes). The Wm2*relu() contraction uses the identity
//              wm2*relu(x) = (wm2/2)*x + (wm2/2)*|x|
//          implemented as float2 v_pk_fma_f32 chains over ADJACENT C-tile
//          element pairs (consecutive VGPRs -> no register marshaling).
//          One xor-16 shuffle combines the meta-row halves; two 16-entry
//          tiles per iteration give one full-wave contiguous 32-entry store.

#define DD 2048
#define MH 32

typedef __attribute__((ext_vector_type(2))) float v2f;
typedef __attribute__((ext_vector_type(8))) float v8f;

__device__ __forceinline__ v2f pair2(const v8f v, int j) {
    v2f r;
    r[0] = v[2 * j];
    r[1] = v[2 * j + 1];
    return r;
}

// ---------------------------------------------------------------------------
// GEMV: out[row] = relu( sum_j W[row][j]*a[j] + b[row] ),  one wave per row.
// ---------------------------------------------------------------------------
__global__ __launch_bounds__(256) void gemv_relu_kernel(
    const float* __restrict__ W, const float* __restrict__ a,
    const float* __restrict__ b, float* __restrict__ out)
{
    const int lane = threadIdx.x & 31;
    const int wave = threadIdx.x >> 5;
    const int row  = blockIdx.x * 8 + wave;

    const float* __restrict__ wrow = W + (size_t)row * DD;

    float acc = 0.0f;
    for (int j = lane * 4; j < DD; j += 32 * 4) {   // 16 b128-load iterations
        const float4 wv = *(const float4*)(wrow + j);
        const float4 av = *(const float4*)(a + j);
        acc += wv.x * av.x + wv.y * av.y + wv.z * av.z + wv.w * av.w;
    }
    acc += __shfl_xor(acc, 1);
    acc += __shfl_xor(acc, 2);
    acc += __shfl_xor(acc, 4);
    acc += __shfl_xor(acc, 8);
    acc += __shfl_xor(acc, 16);
    if (lane == 0) {
        const float v = acc + b[row];
        out[row] = v > 0.0f ? v : 0.0f;
    }
}

// ---------------------------------------------------------------------------
// Meta update: Wout[o][i] = Wm2 . relu( [Wm1;bm1]^T (aPre[i], W[o][i], aPost[o], 1) ) + bm2
// One wave per output row o; 32 entries (two WMMA tiles) per iteration.
// ---------------------------------------------------------------------------
__global__ __launch_bounds__(256) void meta_update_kernel(
    const float* __restrict__ W,    const float* __restrict__ aPre,
    const float* __restrict__ aPost,const float* __restrict__ Wm1,
    const float* __restrict__ bm1,  const float* __restrict__ Wm2,
    const float* __restrict__ bm2,  float* __restrict__ Wout)
{
    __shared__ float sA[DD];
    for (int t = threadIdx.x; t < DD; t += 256) sA[t] = aPre[t];
    __syncthreads();

    const int lane = threadIdx.x & 31;
    const int wave = threadIdx.x >> 5;
    const int col  = lane & 15;   // entry column within tile / meta row for A
    const int half = lane >> 4;   // K half: 0 -> K{0,1}, 1 -> K{2,3}

    // A operands (16 meta rows x 4 K), loop-invariant.
    // Row m: K0=Wm1[0][m], K1=Wm1[1][m], K2=Wm1[2][m], K3=bm1[m].
    // Layout: lanes0-15: VGPR0=K0, VGPR1=K1; lanes16-31: VGPR0=K2, VGPR1=K3.
    v2f A0, A1;   // meta rows 0..15 and 16..31
    A0[0] = half ? Wm1[2 * MH + col]      : Wm1[0 * MH + col];
    A0[1] = half ? bm1[col]               : Wm1[1 * MH + col];
    A1[0] = half ? Wm1[2 * MH + 16 + col] : Wm1[0 * MH + 16 + col];
    A1[1] = half ? bm1[16 + col]          : Wm1[1 * MH + 16 + col];

    // Packed half-weights: wm2*relu(x) = hw*x + hw*|x|, paired over adjacent
    // C-tile elements {2j, 2j+1}. C VGPR r holds meta row (r + 8*half) for
    // tile0 and (16 + r + 8*half) for tile1.
    const float* wmb = Wm2 + (half ? 8 : 0);
    v2f pA[4], pB[4];
#pragma unroll
    for (int j = 0; j < 4; ++j) {
        pA[j][0] = 0.5f * wmb[2 * j];
        pA[j][1] = 0.5f * wmb[2 * j + 1];
        pB[j][0] = 0.5f * wmb[16 + 2 * j];
        pB[j][1] = 0.5f * wmb[16 + 2 * j + 1];
    }

    const float accInit = half ? 0.0f : bm2[0];   // bias folded into one half

    const int row = blockIdx.x * 8 + wave;
    const float* __restrict__ wrow = W    + (size_t)row * DD;
    float*       __restrict__ orow = Wout + (size_t)row * DD;
    const float  vO = aPost[row];

    for (int i0 = 0; i0 < DD; i0 += 32) {
        const float vi0 = sA[i0 + col];
        const float w0  = wrow[i0 + col];
        const float vi1 = sA[i0 + 16 + col];
        const float w1  = wrow[i0 + 16 + col];

        // B operand (4K x 16 entries): K0=v_i, K1=w, K2=v_o, K3=1.
        // Layout: lanes0-15: VGPR0=K0, VGPR1=K1; lanes16-31: VGPR0=K2, VGPR1=K3.
        v2f Ba, Bb;
        Ba[0] = half ? vO   : vi0;
        Ba[1] = half ? 1.0f : w0;
        Bb[0] = half ? vO   : vi1;
        Bb[1] = half ? 1.0f : w1;

        v8f C0a = {}, C1a = {}, C0b = {}, C1b = {};
        C0a = __builtin_amdgcn_wmma_f32_16x16x4_f32(
                  false, A0, false, Ba, (short)0, C0a, false, false);
        C1a = __builtin_amdgcn_wmma_f32_16x16x4_f32(
                  false, A1, false, Ba, (short)0, C1a, false, false);
        C0b = __builtin_amdgcn_wmma_f32_16x16x4_f32(
                  false, A0, false, Bb, (short)0, C0b, false, false);
        C1b = __builtin_amdgcn_wmma_f32_16x16x4_f32(
                  false, A1, false, Bb, (short)0, C1b, false, false);

        // sum_m wm2[m]*relu(x_m) = sum hw*x  +  sum hw*|x|
        // float2 chains over adjacent element pairs -> v_pk_fma_f32 with
        // operands taken directly from consecutive WMMA result VGPRs.
        v2f rawA = {accInit, 0.0f}, absA = {0.0f, 0.0f};
        v2f rawB = {accInit, 0.0f}, absB = {0.0f, 0.0f};
#pragma unroll
        for (int j = 0; j < 4; ++j) {
            const v2f c0a = pair2(C0a, j);
            const v2f c1a = pair2(C1a, j);
            const v2f c0b = pair2(C0b, j);
            const v2f c1b = pair2(C1b, j);
            rawA = __builtin_elementwise_fma(pA[j], c0a, rawA);
            absA = __builtin_elementwise_fma(pA[j], __builtin_elementwise_abs(c0a), absA);
            rawA = __builtin_elementwise_fma(pB[j], c1a, rawA);
            absA = __builtin_elementwise_fma(pB[j], __builtin_elementwise_abs(c1a), absA);
            rawB = __builtin_elementwise_fma(pA[j], c0b, rawB);
            absB = __builtin_elementwise_fma(pA[j], __builtin_elementwise_abs(c0b), absB);
            rawB = __builtin_elementwise_fma(pB[j], c1b, rawB);
            absB = __builtin_elementwise_fma(pB[j], __builtin_elementwise_abs(c1b), absB);
        }
        const v2f tA = rawA + absA;
        const v2f tB = rawB + absB;
        float accA = tA[0] + tA[1];
        float accB = tB[0] + tB[1];

        // combine meta-row halves; afterwards ALL lanes hold the tile result
        accA += __shfl_xor(accA, 16);
        accB += __shfl_xor(accB, 16);

        // lane L < 16 -> entry i0+L (tile a), lane L >= 16 -> entry i0+L (tile b)
        const float res = (lane < 16) ? accA : accB;
        __builtin_nontemporal_store(res, orow + i0 + lane);
    }
}

// ---------------------------------------------------------------------------
// Launch
// ---------------------------------------------------------------------------
extern "C" void kernel_launch(void* const* d_in, const int* in_sizes, int n_in,
                              void* d_out, int out_size, void* d_ws, size_t ws_size,
                              hipStream_t stream) {
    const float* x   = (const float*)d_in[0];
    const float* W1  = (const float*)d_in[1];
    const float* b1  = (const float*)d_in[2];
    const float* W2  = (const float*)d_in[3];
    const float* b2  = (const float*)d_in[4];
    const float* W3  = (const float*)d_in[5];
    const float* b3  = (const float*)d_in[6];
    const float* Wm1 = (const float*)d_in[7];
    const float* bm1 = (const float*)d_in[8];
    const float* Wm2 = (const float*)d_in[9];
    const float* bm2 = (const float*)d_in[10];

    float* out = (float*)d_out;                 // [2048]
    float* nW1 = out + DD;                      // [2048*2048]
    float* nW2 = nW1 + (size_t)DD * DD;
    float* nW3 = nW2 + (size_t)DD * DD;

    float* a1 = (float*)d_ws;                   // [2048]
    float* a2 = a1 + DD;                        // [2048]

    dim3 blk(256);
    dim3 grid(DD / 8);                          // 8 rows (waves) per block

    // Forward (sequential layer dependency)
    gemv_relu_kernel<<<grid, blk, 0, stream>>>(W1, x,  b1, a1);
    gemv_relu_kernel<<<grid, blk, 0, stream>>>(W2, a1, b2, a2);
    gemv_relu_kernel<<<grid, blk, 0, stream>>>(W3, a2, b3, out);

    // Meta updates (read original activations/weights; order independent)
    meta_update_kernel<<<grid, blk, 0, stream>>>(W1, x,  a1,  Wm1, bm1, Wm2, bm2, nW1);
    meta_update_kernel<<<grid, blk, 0, stream>>>(W2, a1, a2,  Wm1, bm1, Wm2, bm2, nW2);
    meta_update_kernel<<<grid, blk, 0, stream>>>(W3, a2, out, Wm1, bm1, Wm2, bm2, nW3);
}